// HyperConnection_60962765799637
// MI455X (gfx1250) — compile-verified
//
#include <hip/hip_runtime.h>
#include <math.h>

typedef float v2f __attribute__((ext_vector_type(2)));
typedef float v4f __attribute__((ext_vector_type(4)));
typedef float v8f __attribute__((ext_vector_type(8)));

static constexpr int S_DIM = 4;
static constexpr int D_DIM = 256;
static constexpr int WAVES = 16;          // samples per block (one wave32 each)
static constexpr int BLOCK = WAVES * 32;  // 512 threads
static constexpr float TAU_INV = 10.0f;   // 1 / 0.1
static constexpr int SINK_ITERS = 10;

__device__ __forceinline__ float fast_tanh(float x) {
#if __has_builtin(__builtin_amdgcn_tanhf)
    return __builtin_amdgcn_tanhf(x);     // v_tanh_f32 (gfx1250 TRANS op)
#else
    // tanh(x) = sign(x) * (1 - 2/(exp(2|x|)+1)); args here are tiny (~alpha*dot)
    const float e = __expf(2.0f * fabsf(x));
    const float r = 1.0f - 2.0f * __builtin_amdgcn_rcpf(e + 1.0f);
    return copysignf(r, x);
#endif
}

__device__ __forceinline__ v4f ld_nt4(const float* p) {
    return __builtin_nontemporal_load(reinterpret_cast<const v4f*>(p));
}
__device__ __forceinline__ void st_nt4(float* p, v4f v) {
    __builtin_nontemporal_store(v, reinterpret_cast<v4f*>(p));
}
__device__ __forceinline__ float sumsq4(const v4f a) {
    return a.x * a.x + a.y * a.y + a.z * a.z + a.w * a.w;
}

__global__ __launch_bounds__(BLOCK) void hyperconn_kernel(
    const float* __restrict__ x,          // N*S*D
    const float* __restrict__ msg,        // N*D
    const float* __restrict__ theta_pre,  // D
    const float* __restrict__ b_pre,      // S
    const float* __restrict__ theta_post, // D
    const float* __restrict__ b_post,     // S
    const float* __restrict__ Theta_res,  // S*D
    const float* __restrict__ B_res,      // S*S
    const float* __restrict__ a_pre,
    const float* __restrict__ a_post,
    const float* __restrict__ a_res,
    float* __restrict__ out_x,            // N*S*D
    float* __restrict__ out_hpre,         // N*S
    int N)
{
    __shared__ float lds_pool[WAVES][D_DIM];  // 16 pooled vectors (A matrix rows)
    __shared__ float lds_W[16][D_DIM];        // 6 weight rows + 10 zero rows (B cols)
    __shared__ float lds_dots[WAVES][8];      // per-sample: [pre, post, res0..3, pad]
    __shared__ float lds_bpre[S_DIM];
    __shared__ float lds_bpost[S_DIM];
    __shared__ float lds_Bres[S_DIM * S_DIM];

    const int tid  = threadIdx.x;
    const int wave = tid >> 5;
    const int lane = tid & 31;
    const int n    = blockIdx.x * WAVES + wave;

    // ---- cooperative weight staging into LDS (reused -> regular temporal) ---
    for (int i = tid; i < D_DIM; i += BLOCK) {
        lds_W[0][i] = theta_pre[i];
        lds_W[1][i] = theta_post[i];
    }
    for (int i = tid; i < S_DIM * D_DIM; i += BLOCK)
        lds_W[2 + (i >> 8)][i & 255] = Theta_res[i];
    for (int i = tid; i < 10 * D_DIM; i += BLOCK)   // pad rows 6..15 with zeros
        lds_W[6 + (i >> 8)][i & 255] = 0.0f;
    if (tid < S_DIM) {
        lds_bpre[tid]  = b_pre[tid];
        lds_bpost[tid] = b_post[tid];
    }
    if (tid < S_DIM * S_DIM) lds_Bres[tid] = B_res[tid];

    const float alpha_pre  = a_pre[0];
    const float alpha_post = a_post[0];
    const float alpha_res  = a_res[0];

    // ---- phase 1: stream x once (NT), row norms, pooled vector --------------
    v4f xa[S_DIM][2];
    if (n < N) {
        const float* xb = x + (size_t)n * (S_DIM * D_DIM) + lane * 8;
        float inv_norm[S_DIM];
        #pragma unroll
        for (int s = 0; s < S_DIM; ++s) {
            xa[s][0] = ld_nt4(xb + s * D_DIM);
            xa[s][1] = ld_nt4(xb + s * D_DIM + 4);
            float ss = sumsq4(xa[s][0]) + sumsq4(xa[s][1]);
            #pragma unroll
            for (int m = 1; m < 32; m <<= 1) ss += __shfl_xor(ss, m, 32);
            inv_norm[s] = __builtin_amdgcn_rcpf(fmaxf(sqrtf(ss), 1e-12f));
        }
        v4f p0 = 0.25f * (xa[0][0] * inv_norm[0] + xa[1][0] * inv_norm[1] +
                          xa[2][0] * inv_norm[2] + xa[3][0] * inv_norm[3]);
        v4f p1 = 0.25f * (xa[0][1] * inv_norm[0] + xa[1][1] * inv_norm[1] +
                          xa[2][1] * inv_norm[2] + xa[3][1] * inv_norm[3]);
        *reinterpret_cast<v4f*>(&lds_pool[wave][lane * 8])     = p0;
        *reinterpret_cast<v4f*>(&lds_pool[wave][lane * 8 + 4]) = p1;
    }
    __syncthreads();

    // ---- phase 2 (wave 0): dots = pooled(16xD) x W^T(Dx16) via WMMA ---------
    // V_WMMA_F32_16X16X4_F32 layout (ISA 7.12.2, 32-bit A 16x4):
    //   lanes 0-15 : M/N = lane,    VGPR0 = K+0, VGPR1 = K+1
    //   lanes 16-31: M/N = lane-16, VGPR0 = K+2, VGPR1 = K+3
    if (wave == 0) {
        const int row  = lane & 15;
        const int koff = (lane >> 4) << 1;
        const float* prow = &lds_pool[row][0];
        const float* wrow = &lds_W[row][0];
        v8f acc = {};
        #pragma unroll
        for (int c = 0; c < D_DIM; c += 4) {
            v2f a = *reinterpret_cast<const v2f*>(prow + c + koff);
            v2f b = *reinterpret_cast<const v2f*>(wrow + c + koff);
            acc = __builtin_amdgcn_wmma_f32_16x16x4_f32(
                false, a, false, b, (short)0, acc, false, false);
        }
        // C layout: VGPR v -> row v (lanes 0-15) / row v+8 (lanes 16-31), col=lane&15
        if (row < 6) {
            const int rbase = (lane >> 4) << 3;
            #pragma unroll
            for (int v = 0; v < 8; ++v) lds_dots[rbase + v][row] = acc[v];
        }
    }
    __syncthreads();

    // ---- phase 3: gates, Sinkhorn, residual bmm + rank-1 message add --------
    if (n < N) {
        const float d_pre  = lds_dots[wave][0];
        const float d_post = lds_dots[wave][1];

        // H_pre output: alpha_pre * tanh(pooled . theta_pre) + b_pre
        const float tpre = fast_tanh(d_pre);
        if (lane < S_DIM)
            out_hpre[(size_t)n * S_DIM + lane] = alpha_pre * tpre + lds_bpre[lane];

        const float tpost = fast_tanh(d_post);
        float hp[S_DIM];
        #pragma unroll
        for (int s = 0; s < S_DIM; ++s) hp[s] = alpha_post * tpost + lds_bpost[s];

        // Sinkhorn on 4x4, distributed: lane (mod 16) owns element (s,t)
        const int es = (lane >> 2) & 3;
        const int et = lane & 3;
        float mv = __expf((alpha_res * fast_tanh(lds_dots[wave][2 + es]) +
                           lds_Bres[es * 4 + et]) * TAU_INV);
        #pragma unroll
        for (int it = 0; it < SINK_ITERS; ++it) {
            float r = mv;                       // row sum over t (quad)
            r += __shfl_xor(r, 1, 32);
            r += __shfl_xor(r, 2, 32);
            mv *= __builtin_amdgcn_rcpf(r);
            float cc = mv;                      // col sum over s
            cc += __shfl_xor(cc, 4, 32);
            cc += __shfl_xor(cc, 8, 32);
            mv *= __builtin_amdgcn_rcpf(cc);
        }
        float M[S_DIM][S_DIM];
        #pragma unroll
        for (int s = 0; s < S_DIM; ++s)
            #pragma unroll
            for (int t = 0; t < S_DIM; ++t)
                M[s][t] = __shfl(mv, s * 4 + t, 32);

        const float* mb = msg + (size_t)n * D_DIM + lane * 8;
        v4f g0 = ld_nt4(mb);
        v4f g1 = ld_nt4(mb + 4);

        float* ob = out_x + (size_t)n * (S_DIM * D_DIM) + lane * 8;
        #pragma unroll
        for (int s = 0; s < S_DIM; ++s) {
            v4f o0 = M[s][0] * xa[0][0] + M[s][1] * xa[1][0] +
                     M[s][2] * xa[2][0] + M[s][3] * xa[3][0] + hp[s] * g0;
            v4f o1 = M[s][0] * xa[0][1] + M[s][1] * xa[1][1] +
                     M[s][2] * xa[2][1] + M[s][3] * xa[3][1] + hp[s] * g1;
            st_nt4(ob + s * D_DIM,     o0);
            st_nt4(ob + s * D_DIM + 4, o1);
        }
    }
}

extern "C" void kernel_launch(void* const* d_in, const int* in_sizes, int n_in,
                              void* d_out, int out_size, void* d_ws, size_t ws_size,
                              hipStream_t stream) {
    (void)n_in; (void)d_ws; (void)ws_size; (void)out_size;
    const float* x          = (const float*)d_in[0];
    const float* msg        = (const float*)d_in[1];
    const float* theta_pre  = (const float*)d_in[2];
    const float* b_pre      = (const float*)d_in[3];
    const float* theta_post = (const float*)d_in[4];
    const float* b_post     = (const float*)d_in[5];
    const float* Theta_res  = (const float*)d_in[6];
    const float* B_res      = (const float*)d_in[7];
    const float* a_pre      = (const float*)d_in[8];
    const float* a_post     = (const float*)d_in[9];
    const float* a_res      = (const float*)d_in[10];

    const int N = in_sizes[0] / (S_DIM * D_DIM);
    float* out_x    = (float*)d_out;
    float* out_hpre = out_x + (size_t)N * S_DIM * D_DIM;

    const int blocks = (N + WAVES - 1) / WAVES;
    hyperconn_kernel<<<blocks, BLOCK, 0, stream>>>(
        x, msg, theta_pre, b_pre, theta_post, b_post, Theta_res, B_res,
        a_pre, a_post, a_res, out_x, out_hpre, N);
}